// GraphNet_30116310679885
// MI455X (gfx1250) — compile-verified
//
#include <hip/hip_runtime.h>

#define HID     128
#define NOUTF   64
#define NGRAPH  256

typedef __attribute__((ext_vector_type(2))) float v2f;
typedef __attribute__((ext_vector_type(8))) float v8f;

// ---------------- degree / norm ----------------
__global__ void k_deg_init(float* deg, int n) {
  int i = blockIdx.x * blockDim.x + threadIdx.x;
  if (i < n) deg[i] = 1.0f;                      // self-loop contributes 1
}

__global__ void k_deg_count(const int* __restrict__ row, float* deg, int e) {
  int i = blockIdx.x * blockDim.x + threadIdx.x;
  if (i < e) atomicAdd(&deg[row[i]], 1.0f);
}

__global__ void k_dinv(float* deg, int n) {
  int i = blockIdx.x * blockDim.x + threadIdx.x;
  if (i < n) deg[i] = rsqrtf(deg[i]);            // deg >= 1 always (self-loop)
}

// ---------------- embedding gather ----------------
__global__ void k_embed(const int* __restrict__ H, const float* __restrict__ emb,
                        float* __restrict__ X, int n) {
  int i = blockIdx.x * blockDim.x + threadIdx.x;
  if (i < n * HID) {
    int node = i >> 7;
    int f    = i & (HID - 1);
    X[i] = emb[H[node] * HID + f];
  }
}

// ---------------- WMMA GEMM: Y[M x nOut] = relu?(X)[M x 128] @ W[nOut x 128]^T + bias ----
// blockDim.x = (nOut/16)*32, gridDim.x = M/16. One wave per 16x16 output tile.
// The 16x128 A tile is staged once per block into LDS (ReLU fused there);
// each wave then feeds its 32 x V_WMMA_F32_16X16X4_F32 from ds_load_b64.
// f32 16x16x4 fragment layout: lane L<16 holds row (L&15), K pair {0,1};
// lane L>=16 holds row (L&15), K pair {2,3}. B rows (W) mirror this addressing.
template <int RELU>
__global__ void k_gemm_wmma(const float* __restrict__ X, const float* __restrict__ W,
                            const float* __restrict__ bias, float* __restrict__ Y,
                            int nOut) {
  __shared__ float As[16 * HID];                 // 8 KB

  const int tid   = threadIdx.x;
  const int tileM = blockIdx.x << 4;

  // cooperative stage of A tile (2048 floats) as float4, ReLU applied once here
  for (int i = tid; i < (16 * HID) / 4; i += blockDim.x) {
    float4 v = *(const float4*)(X + (size_t)tileM * HID + i * 4);
    if (RELU) {
      v.x = fmaxf(v.x, 0.f); v.y = fmaxf(v.y, 0.f);
      v.z = fmaxf(v.z, 0.f); v.w = fmaxf(v.w, 0.f);
    }
    *(float4*)(As + i * 4) = v;
  }
  __syncthreads();

  const int wave  = tid >> 5;
  const int lane  = tid & 31;
  const int r     = lane & 15;
  const int kph   = (lane >> 4) << 1;            // 0 for lanes 0-15, 2 for lanes 16-31
  const int tileN = wave << 4;

  const float* __restrict__ arow = As + r * HID + kph;
  const float* __restrict__ brow = W + (size_t)(tileN + r) * HID + kph;

  v8f acc = {0.f, 0.f, 0.f, 0.f, 0.f, 0.f, 0.f, 0.f};
#pragma unroll
  for (int k0 = 0; k0 < HID; k0 += 4) {
    v2f a = *(const v2f*)(arow + k0);            // ds_load_b64
    v2f b = *(const v2f*)(brow + k0);            // global_load_b64 (W is hot in L0)
    // 8 args: (neg_a, A, neg_b, B, c_mod, C, reuse_a, reuse_b)
    acc = __builtin_amdgcn_wmma_f32_16x16x4_f32(false, a, false, b, (short)0, acc,
                                                false, false);
  }

  // D layout: lanes 0-15 -> M = tileM + v, N = lane; lanes 16-31 -> M = tileM+8+v, N = lane-16
  const int   col   = tileN + r;
  const float bb    = bias[col];
  const int   mbase = tileM + ((lane < 16) ? 0 : 8);
#pragma unroll
  for (int v = 0; v < 8; ++v) {
    Y[(size_t)(mbase + v) * nOut + col] = acc[v] + bb;
  }
}

// ---------------- aggregation: self-loop init then edge scatter ----------------
__global__ void k_self_init(const float* __restrict__ h, const float* __restrict__ dinv,
                            float* __restrict__ agg, int n) {
  int i = blockIdx.x * blockDim.x + threadIdx.x;
  if (i < n * HID) {
    int node = i >> 7;
    float d = dinv[node];
    agg[i] = h[i] * d * d;                       // self edge: norm = dinv[n]^2
  }
}

// one wave per edge, 4 features per lane
__global__ void k_edge_scatter(const int* __restrict__ row, const int* __restrict__ col,
                               const float* __restrict__ dinv, const float* __restrict__ h,
                               float* __restrict__ agg, int e) {
  int eid = blockIdx.x * (blockDim.x >> 5) + (threadIdx.x >> 5);
  if (eid >= e) return;
  int lane = threadIdx.x & 31;
  int rr = row[eid];
  int cc = col[eid];
  float nrm = dinv[rr] * dinv[cc];
  const float4 hv = *(const float4*)(h + (size_t)cc * HID + lane * 4);
  float* out = agg + (size_t)rr * HID + lane * 4;
  atomicAdd(out + 0, hv.x * nrm);
  atomicAdd(out + 1, hv.y * nrm);
  atomicAdd(out + 2, hv.z * nrm);
  atomicAdd(out + 3, hv.w * nrm);
}

// ---------------- mean pool (ReLU fused on read) ----------------
__global__ void k_pool_zero(float* pooled, float* cnt) {
  int i = blockIdx.x * blockDim.x + threadIdx.x;
  if (i < NGRAPH * HID) pooled[i] = 0.f;
  if (i < NGRAPH) cnt[i] = 0.f;
}

__global__ void k_pool_accum(const int* __restrict__ batch, const float* __restrict__ x,
                             float* pooled, float* cnt, int n) {
  int node = blockIdx.x * (blockDim.x >> 5) + (threadIdx.x >> 5);
  if (node >= n) return;
  int lane = threadIdx.x & 31;
  int g = batch[node];
  const float4 xv = *(const float4*)(x + (size_t)node * HID + lane * 4);
  float* out = pooled + (size_t)g * HID + lane * 4;
  atomicAdd(out + 0, fmaxf(xv.x, 0.f));
  atomicAdd(out + 1, fmaxf(xv.y, 0.f));
  atomicAdd(out + 2, fmaxf(xv.z, 0.f));
  atomicAdd(out + 3, fmaxf(xv.w, 0.f));
  if (lane == 0) atomicAdd(&cnt[g], 1.0f);
}

__global__ void k_pool_div(float* pooled, const float* __restrict__ cnt) {
  int i = blockIdx.x * blockDim.x + threadIdx.x;
  if (i < NGRAPH * HID) pooled[i] /= fmaxf(cnt[i >> 7], 1.0f);
}

// ---------------- driver ----------------
extern "C" void kernel_launch(void* const* d_in, const int* in_sizes, int n_in,
                              void* d_out, int out_size, void* d_ws, size_t ws_size,
                              hipStream_t stream) {
  const int*   H     = (const int*)d_in[0];
  const int*   ei    = (const int*)d_in[1];
  const int*   batch = (const int*)d_in[2];
  const float* emb   = (const float*)d_in[3];
  const float* W1    = (const float*)d_in[4];
  const float* b1    = (const float*)d_in[5];
  const float* W2    = (const float*)d_in[6];
  const float* b2    = (const float*)d_in[7];
  const float* Wout  = (const float*)d_in[8];
  const float* bout  = (const float*)d_in[9];
  float*       out   = (float*)d_out;

  const int N = in_sizes[0];          // 50000 nodes
  const int E = in_sizes[1] / 2;      // 600000 edges
  const int* row = ei;                // edge_index[0, :]
  const int* col = ei + E;            // edge_index[1, :]

  // workspace carve-up (all 256B aligned; total ~51.6 MB)
  char*  ws  = (char*)d_ws;
  size_t NB  = (size_t)N * HID * sizeof(float);          // 25.6 MB
  NB = (NB + 255) & ~(size_t)255;
  size_t DB  = ((size_t)N * sizeof(float) + 255) & ~(size_t)255;
  float* bufA   = (float*)(ws);
  float* bufB   = (float*)(ws + NB);
  float* dinv   = (float*)(ws + 2 * NB);
  float* pooled = (float*)(ws + 2 * NB + DB);
  float* cnt    = (float*)(ws + 2 * NB + DB + (size_t)NGRAPH * HID * sizeof(float));

  const int T = 256;
  const int nfBlocks = (N * HID + T - 1) / T;

  // degrees -> dinv (stored in `dinv`)
  k_deg_init <<<(N + T - 1) / T, T, 0, stream>>>(dinv, N);
  k_deg_count<<<(E + T - 1) / T, T, 0, stream>>>(row, dinv, E);
  k_dinv     <<<(N + T - 1) / T, T, 0, stream>>>(dinv, N);

  // x0 = emb[H]
  k_embed<<<nfBlocks, T, 0, stream>>>(H, emb, bufA, N);

  // layer 1: h1 = x0 @ W1^T + b1 ; agg1 = scatter(h1)
  k_gemm_wmma<0><<<N / 16, 256, 0, stream>>>(bufA, W1, b1, bufB, HID);
  k_self_init   <<<nfBlocks, T, 0, stream>>>(bufB, dinv, bufA, N);
  k_edge_scatter<<<(E + 7) / 8, 256, 0, stream>>>(row, col, dinv, bufB, bufA, E);

  // layer 2: h2 = relu(agg1) @ W2^T + b2 ; agg2 = scatter(h2)
  k_gemm_wmma<1><<<N / 16, 256, 0, stream>>>(bufA, W2, b2, bufB, HID);
  k_self_init   <<<nfBlocks, T, 0, stream>>>(bufB, dinv, bufA, N);
  k_edge_scatter<<<(E + 7) / 8, 256, 0, stream>>>(row, col, dinv, bufB, bufA, E);

  // mean pool over relu(agg2)
  k_pool_zero <<<(NGRAPH * HID + T - 1) / T, T, 0, stream>>>(pooled, cnt);
  k_pool_accum<<<(N + 7) / 8, 256, 0, stream>>>(batch, bufA, pooled, cnt, N);
  k_pool_div  <<<(NGRAPH * HID + T - 1) / T, T, 0, stream>>>(pooled, cnt);

  // head: out = pooled @ Wout^T + bout   (256x128 @ 128x64)
  k_gemm_wmma<0><<<NGRAPH / 16, (NOUTF / 16) * 32, 0, stream>>>(pooled, Wout, bout, out, NOUTF);
}